// GCN_SSE_86878598463961
// MI455X (gfx1250) — compile-verified
//
#include <hip/hip_runtime.h>

typedef __attribute__((ext_vector_type(2))) float v2f;
typedef __attribute__((ext_vector_type(8))) float v8f;

#define ALPHA_F 0.1f

// ---------------- utility kernels ----------------

__global__ __launch_bounds__(256) void k_zero(float* __restrict__ p, long n) {
  long i = (long)blockIdx.x * blockDim.x + threadIdx.x;
  long stride = (long)gridDim.x * blockDim.x;
  for (; i < n; i += stride) p[i] = 0.0f;
}

__global__ __launch_bounds__(256) void k_degree(const long long* __restrict__ src,
                                                const long long* __restrict__ dst,
                                                float* __restrict__ deg_out,
                                                float* __restrict__ deg_in, int E) {
  int i = blockIdx.x * blockDim.x + threadIdx.x;
  if (i < E) {
    atomicAdd(&deg_out[(int)src[i]], 1.0f);
    atomicAdd(&deg_in[(int)dst[i]], 1.0f);
  }
}

__global__ __launch_bounds__(256) void k_norm(float* __restrict__ p, int n) {
  int i = blockIdx.x * blockDim.x + threadIdx.x;
  if (i < n) {
    float d = p[i];
    p[i] = rsqrtf(d < 1.0f ? 1.0f : d);
  }
}

// ---------------- WMMA fp32 GEMM: H = (X @ W) * norm_out[:,None] ----------------
// One wave computes one 16x16 output tile with V_WMMA_F32_16X16X4_F32.
// K, F are compile-time so all A/B loads use immediate offsets from two base
// pointers, and the K-loop fully unrolls. Two independent accumulator chains
// (K split in halves) give WMMA ILP; summed at the end.
// A-layout (16x4 f32): lanes 0-15 hold M=lane, K={0,1}; lanes 16-31 K={2,3}.
// B-layout (4x16 f32): lanes 0-15 hold N=lane, K={0,1}; lanes 16-31 K={2,3}.
// C/D (16x16 f32, 8 VGPRs): vgpr r: lanes 0-15 -> (M=r,N=lane); 16-31 -> (M=r+8,N=lane-16).
template <int K, int F>
__global__ __launch_bounds__(256)
void k_gemm_scale(const float* __restrict__ X, const float* __restrict__ W,
                  const float* __restrict__ norm_out, float* __restrict__ H,
                  int N) {
  int gw   = (blockIdx.x * blockDim.x + threadIdx.x) >> 5;
  int lane = threadIdx.x & 31;
  constexpr int tilesN = F >> 4;
  int tileM = gw / tilesN;
  int tileN = gw - tileM * tilesN;
  if ((tileM << 4) >= N) return;            // wave-uniform guard (EXEC all-1 inside)

  int rowBase = tileM << 4;
  int colBase = tileN << 4;
  int l15 = lane & 15;
  int hi  = lane >> 4;                      // 0: lanes 0-15, 1: lanes 16-31

  // lane-fixed base pointers; all loop offsets are compile-time immediates
  const float* xrow = X + (size_t)(rowBase + l15) * K + (hi << 1);
  const float* wcol = W + (size_t)(hi << 1) * F + colBase + l15;

  v8f acc0 = {0.f, 0.f, 0.f, 0.f, 0.f, 0.f, 0.f, 0.f};
  v8f acc1 = {0.f, 0.f, 0.f, 0.f, 0.f, 0.f, 0.f, 0.f};
  constexpr int KH = K / 2;

#pragma unroll
  for (int k = 0; k < KH; k += 4) {
    {
      v2f a = *(const v2f*)(xrow + k);
      v2f b;
      b.x = wcol[k * F];
      b.y = wcol[k * F + F];
      acc0 = __builtin_amdgcn_wmma_f32_16x16x4_f32(false, a, false, b,
                                                   (short)0, acc0, false, false);
    }
    {
      v2f a = *(const v2f*)(xrow + KH + k);
      v2f b;
      b.x = wcol[(KH + k) * F];
      b.y = wcol[(KH + k) * F + F];
      acc1 = __builtin_amdgcn_wmma_f32_16x16x4_f32(false, a, false, b,
                                                   (short)0, acc1, false, false);
    }
  }

  int col = colBase + l15;
  float* hp = H + (size_t)(rowBase + (hi << 3)) * F + col;
#pragma unroll
  for (int r = 0; r < 8; ++r) {
    int row = rowBase + (hi << 3) + r;
    hp[(size_t)r * F] = (acc0[r] + acc1[r]) * norm_out[row];
  }
}

// ---------------- edge scatter: agg[dst] += h[src] ----------------
// one wave per edge; 128 feats = 32 lanes x float4
__global__ __launch_bounds__(256)
void k_scatter128(const float* __restrict__ h, const long long* __restrict__ src,
                  const long long* __restrict__ dst, float* __restrict__ agg, int E) {
  int gtid = blockIdx.x * blockDim.x + threadIdx.x;
  int wave = gtid >> 5;
  int lane = gtid & 31;
  int nw = (gridDim.x * blockDim.x) >> 5;
  for (int e = wave; e < E; e += nw) {
    long s = (long)src[e];
    long d = (long)dst[e];
    float4 v = *((const float4*)(h + s * 128) + lane);
    float* ap = agg + d * 128 + lane * 4;
    atomicAdd(ap + 0, v.x);
    atomicAdd(ap + 1, v.y);
    atomicAdd(ap + 2, v.z);
    atomicAdd(ap + 3, v.w);
  }
}

// 64 feats = 32 lanes x float2
__global__ __launch_bounds__(256)
void k_scatter64(const float* __restrict__ h, const long long* __restrict__ src,
                 const long long* __restrict__ dst, float* __restrict__ agg, int E) {
  int gtid = blockIdx.x * blockDim.x + threadIdx.x;
  int wave = gtid >> 5;
  int lane = gtid & 31;
  int nw = (gridDim.x * blockDim.x) >> 5;
  for (int e = wave; e < E; e += nw) {
    long s = (long)src[e];
    long d = (long)dst[e];
    float2 v = *((const float2*)(h + s * 64) + lane);
    float* ap = agg + d * 64 + lane * 2;
    atomicAdd(ap + 0, v.x);
    atomicAdd(ap + 1, v.y);
  }
}

// ---------------- finalize: out = [blend with] relu?(agg*norm_in + b) ----------------
// f4shift: log2(float4s per row) = 5 for F=128, 4 for F=64
__global__ __launch_bounds__(256)
void k_finalize(const float* __restrict__ agg, const float* __restrict__ norm_in,
                const float* __restrict__ bias, const float* __restrict__ blend_src,
                float alpha, int do_relu, float* __restrict__ out,
                long nrows, int f4shift) {
  long i = (long)blockIdx.x * blockDim.x + threadIdx.x;
  long total = nrows << f4shift;
  long stride = (long)gridDim.x * blockDim.x;
  long mask = (1L << f4shift) - 1;
  for (; i < total; i += stride) {
    long row = i >> f4shift;
    int c4 = (int)(i & mask);
    float ni = norm_in[row];
    float4 a = ((const float4*)agg)[i];
    float4 bb = ((const float4*)bias)[c4];
    float vx = a.x * ni + bb.x;
    float vy = a.y * ni + bb.y;
    float vz = a.z * ni + bb.z;
    float vw = a.w * ni + bb.w;
    if (do_relu) {
      vx = fmaxf(vx, 0.f); vy = fmaxf(vy, 0.f);
      vz = fmaxf(vz, 0.f); vw = fmaxf(vw, 0.f);
    }
    if (blend_src) {
      float4 h = ((const float4*)blend_src)[i];
      vx = (1.f - alpha) * h.x + alpha * vx;
      vy = (1.f - alpha) * h.y + alpha * vy;
      vz = (1.f - alpha) * h.z + alpha * vz;
      vw = (1.f - alpha) * h.w + alpha * vw;
    }
    float4 r; r.x = vx; r.y = vy; r.z = vz; r.w = vw;
    ((float4*)out)[i] = r;
  }
}

// ---------------- host driver ----------------

extern "C" void kernel_launch(void* const* d_in, const int* in_sizes, int n_in,
                              void* d_out, int out_size, void* d_ws, size_t ws_size,
                              hipStream_t stream) {
  const float*     x   = (const float*)d_in[0];
  const long long* src = (const long long*)d_in[1];
  const long long* dst = (const long long*)d_in[2];
  const float* W1 = (const float*)d_in[3];
  const float* b1 = (const float*)d_in[4];
  const float* W2 = (const float*)d_in[5];
  const float* b2 = (const float*)d_in[6];
  const float* W3 = (const float*)d_in[7];
  const float* b3 = (const float*)d_in[8];

  const int IN_F = 128;
  const long N = (long)in_sizes[0] / IN_F;
  const int  E = in_sizes[1];

  float* ws       = (float*)d_ws;
  float* norm_out = ws;                     // N (deg_out -> norm_out in place)
  float* norm_in  = ws + N;                 // N
  float* T        = ws + 2 * N;             // 128*N  (GEMM output / later H2u)
  float* AGG      = ws + 2 * N + 128 * N;   // 128*N
  float* H1       = ws + 2 * N + 256 * N;   // 128*N  (later reused as T3 [N,64])

  // degrees + norms (graph-only, recomputed each call for determinism)
  k_zero<<<2048, 256, 0, stream>>>(norm_out, 2 * N);
  k_degree<<<(E + 255) / 256, 256, 0, stream>>>(src, dst, norm_out, norm_in, E);
  k_norm<<<((int)(2 * N) + 255) / 256, 256, 0, stream>>>(norm_out, (int)(2 * N));

  long tilesM = N / 16;
  int blocks128 = (int)((tilesM * 8 * 32 + 255) / 256);   // F=128 -> 8 col tiles
  int blocks64  = (int)((tilesM * 4 * 32 + 255) / 256);   // F=64  -> 4 col tiles

  // ---- Layer 1: H1 = relu(conv(x, W1, b1)) ----
  k_gemm_scale<128, 128><<<blocks128, 256, 0, stream>>>(x, W1, norm_out, T, (int)N);
  k_zero<<<4096, 256, 0, stream>>>(AGG, 128 * N);
  k_scatter128<<<8192, 256, 0, stream>>>(T, src, dst, AGG, E);
  k_finalize<<<8192, 256, 0, stream>>>(AGG, norm_in, b1, nullptr, 0.f, 1, H1, N, 5);

  // ---- Layer 2: H2u = 0.9*H1 + 0.1*relu(conv(H1, W2, b2)) -> stored in T ----
  k_gemm_scale<128, 128><<<blocks128, 256, 0, stream>>>(H1, W2, norm_out, T, (int)N);
  k_zero<<<4096, 256, 0, stream>>>(AGG, 128 * N);
  k_scatter128<<<8192, 256, 0, stream>>>(T, src, dst, AGG, E);
  k_finalize<<<8192, 256, 0, stream>>>(AGG, norm_in, b2, H1, ALPHA_F, 1, T, N, 5);

  // ---- Layer 3: out = conv(H2u, W3, b3) (no relu) ----
  k_gemm_scale<128, 64><<<blocks64, 256, 0, stream>>>(T, W3, norm_out, H1, (int)N);
  k_zero<<<4096, 256, 0, stream>>>(AGG, 64 * N);
  k_scatter64<<<8192, 256, 0, stream>>>(H1, src, dst, AGG, E);
  k_finalize<<<8192, 256, 0, stream>>>(AGG, norm_in, b3, nullptr, 0.f, 0,
                                       (float*)d_out, N, 4);
}